// MixPool_10703058501944
// MI455X (gfx1250) — compile-verified
//
#include <hip/hip_runtime.h>
#include <cstdint>

#define NN    524288
#define CC    128
#define NG    64
#define NS    8192
#define EPSV  1e-5f
#define NEG_INF (-__builtin_huge_valf())

typedef float v2f __attribute__((ext_vector_type(2)));
typedef float v8f __attribute__((ext_vector_type(8)));

// order-preserving float <-> uint transform for atomicMax-based float max
__device__ __forceinline__ unsigned enc32(float f) {
    unsigned u = __float_as_uint(f);
    return (u & 0x80000000u) ? ~u : (u | 0x80000000u);
}
__device__ __forceinline__ float dec32(unsigned k) {
    unsigned u = (k & 0x80000000u) ? (k & 0x7FFFFFFFu) : ~k;
    return __uint_as_float(u);
}

// ---------------- kernel 1: zero encoded max tables ----------------
__global__ void k_zero(unsigned* __restrict__ p, int n) {
    int i = blockIdx.x * 256 + threadIdx.x;
    if (i < n) p[i] = 0u;   // 0 < enc32(x) for every finite x
}

// ---------------- kernel 2: pack W into k-pair-interleaved float2 ----------------
// Wp[m][(k/2)*128 + n] = { W[k][n], W[k+1][n] }  (k even)
__global__ void k_pack(const float* __restrict__ Wmax, const float* __restrict__ Wsk,
                       float2* __restrict__ Wp) {
    int i = blockIdx.x * 256 + threadIdx.x;      // 0 .. 16383
    int m  = i >> 13;
    int r  = i & 8191;
    int kp = r >> 7;
    int n  = r & 127;
    const float* W = m ? Wsk : Wmax;
    Wp[i] = make_float2(W[(2 * kp) * 128 + n], W[(2 * kp + 1) * 128 + n]);
}

// ---------------- kernel 3: WMMA GEMM + column stats + sorted segment-max ----------------
// block = 128 rows x 128 cols, 8 waves, each wave 16 rows, both weight matrices.
__global__ __launch_bounds__(256)
void k_gemm(const float* __restrict__ x,
            const float2* __restrict__ Wp,          // [2][64*128] packed
            const int* __restrict__ batch,
            const int* __restrict__ stroke,
            float* __restrict__ partials,           // [4096][2][2][128]
            unsigned* __restrict__ g_enc,           // [64][128]
            unsigned* __restrict__ s_enc)           // [8192][128]
{
    __shared__ float htile[128 * 129];
    __shared__ float scratch[256];
    __shared__ int   idbuf[2][128];

    const int tid  = threadIdx.x;
    const int wave = tid >> 5;
    const int lane = tid & 31;
    const int r0   = blockIdx.x * 128;

    if (tid < 128) idbuf[0][tid]       = batch[r0 + tid];
    else           idbuf[1][tid - 128] = stroke[r0 + tid - 128];

    const int m16 = lane & 15;          // M (A) or N (B) within tile
    const int hh  = lane >> 4;          // half selects K pair
    const float* xrow = x + (size_t)(r0 + wave * 16 + m16) * 128 + 2 * hh;
    const float2* Wp0 = Wp;
    const float2* Wp1 = Wp + 64 * 128;

    v8f acc[2][8];
    const v8f vz = {0.f, 0.f, 0.f, 0.f, 0.f, 0.f, 0.f, 0.f};
#pragma unroll
    for (int m = 0; m < 2; ++m)
#pragma unroll
        for (int ct = 0; ct < 8; ++ct) acc[m][ct] = vz;

    for (int kk = 0; kk < 32; ++kk) {             // K = 4*kk
        v2f a = *(const v2f*)(xrow + kk * 4);     // x[row][4kk+2h .. +1]
        const float2* bp0 = Wp0 + (2 * kk + hh) * 128 + m16;
        const float2* bp1 = Wp1 + (2 * kk + hh) * 128 + m16;
#pragma unroll
        for (int ct = 0; ct < 8; ++ct) {
            float2 b0f = bp0[ct * 16];
            v2f b0 = {b0f.x, b0f.y};
            acc[0][ct] = __builtin_amdgcn_wmma_f32_16x16x4_f32(
                false, a, false, b0, (short)0, acc[0][ct], false, false);
            float2 b1f = bp1[ct * 16];
            v2f b1 = {b1f.x, b1f.y};
            acc[1][ct] = __builtin_amdgcn_wmma_f32_16x16x4_f32(
                false, a, false, b1, (short)0, acc[1][ct], false, false);
        }
    }

    // per matrix: stage tile in LDS, reduce stats, run-length segment-max
    for (int m = 0; m < 2; ++m) {
        __syncthreads();
#pragma unroll
        for (int ct = 0; ct < 8; ++ct)
#pragma unroll
            for (int r = 0; r < 8; ++r)
                htile[(wave * 16 + r + 8 * hh) * 129 + ct * 16 + m16] = acc[m][ct][r];
        __syncthreads();

        const int col  = tid & 127;
        const int half = tid >> 7;
        unsigned* table = (m == 0) ? g_enc : s_enc;

        float sum = 0.f, sq = 0.f;
        int   cur    = idbuf[m][half * 64];
        float runmax = NEG_INF;
        for (int r = half * 64; r < half * 64 + 64; ++r) {
            float v = htile[r * 129 + col];
            sum += v; sq += v * v;
            int id = idbuf[m][r];
            if (id != cur) {
                atomicMax(table + (size_t)cur * 128 + col, enc32(runmax));
                cur = id; runmax = v;
            } else {
                runmax = fmaxf(runmax, v);
            }
        }
        atomicMax(table + (size_t)cur * 128 + col, enc32(runmax));

        if (half == 1) { scratch[col] = sum; scratch[128 + col] = sq; }
        __syncthreads();
        if (half == 0) {
            sum += scratch[col]; sq += scratch[128 + col];
            size_t base = (((size_t)blockIdx.x * 2 + m) * 2) * 128 + col;
            partials[base]       = sum;
            partials[base + 128] = sq;
        }
    }
}

// ---------------- kernel 4: reduce per-block stat partials ----------------
__global__ void k_reduce(const float* __restrict__ partials, float* __restrict__ reduced) {
    __shared__ float sh0[256], sh1[256];
    const int t   = threadIdx.x;
    const int m   = blockIdx.x >> 7;
    const int col = blockIdx.x & 127;
    float s = 0.f, q = 0.f;
    for (int i = t; i < 4096; i += 256) {
        size_t base = (((size_t)i * 2 + m) * 2) * 128 + col;
        s += partials[base];
        q += partials[base + 128];
    }
    sh0[t] = s; sh1[t] = q;
    __syncthreads();
    for (int off = 128; off; off >>= 1) {
        if (t < off) { sh0[t] += sh0[t + off]; sh1[t] += sh1[t + off]; }
        __syncthreads();
    }
    if (t == 0) {
        reduced[(m * 2 + 0) * 128 + col] = sh0[0];
        reduced[(m * 2 + 1) * 128 + col] = sh1[0];
    }
}

// ---------------- kernel 5: decode tables, apply BN affine + ReLU ----------------
__global__ void k_finalize(const unsigned* __restrict__ g_enc, const unsigned* __restrict__ s_enc,
                           const float* __restrict__ reduced,
                           const float* __restrict__ gmax, const float* __restrict__ bemax,
                           const float* __restrict__ gsk,  const float* __restrict__ besk,
                           float* __restrict__ gtab, float* __restrict__ stab) {
    int idx = blockIdx.x * 256 + threadIdx.x;   // < (NG+NS)*128
    float sum, sq, ga, be, v;
    float* dst;
    if (idx < NG * CC) {
        int col = idx & 127;
        sum = reduced[col];       sq = reduced[128 + col];
        ga  = gmax[col];          be = bemax[col];
        v   = dec32(g_enc[idx]);  dst = gtab + idx;
    } else {
        int j = idx - NG * CC;
        int col = j & 127;
        sum = reduced[256 + col]; sq = reduced[384 + col];
        ga  = gsk[col];           be = besk[col];
        v   = dec32(s_enc[j]);    dst = stab + j;
    }
    const float inv = 1.f / (float)NN;
    float mu  = sum * inv;
    float var = sq * inv - mu * mu;
    float sc  = ga * rsqrtf(var + EPSV);
    float sh  = be - mu * sc;
    *dst = fmaxf(v * sc + sh, 0.f);
}

// ---------------- kernel 6: gather/broadcast to output ----------------
// out[n][0:128] = stab[stroke[n]], out[n][128:256] = gtab[batch[n]]
__global__ void k_gather(const int* __restrict__ batch, const int* __restrict__ stroke,
                         const float* __restrict__ gtab, const float* __restrict__ stab,
                         float* __restrict__ out) {
    int id = blockIdx.x * 256 + threadIdx.x;    // N*64 float4 slots
    int n  = id >> 6;
    int q  = id & 63;
    float4* outrow = (float4*)(out + (size_t)n * 256);
    if (q < 32) {
        int s = stroke[n];
        outrow[q] = ((const float4*)(stab + (size_t)s * 128))[q];
    } else {
        int g = batch[n];
        outrow[q] = ((const float4*)(gtab + (size_t)g * 128))[q - 32];
    }
}

extern "C" void kernel_launch(void* const* d_in, const int* in_sizes, int n_in,
                              void* d_out, int out_size, void* d_ws, size_t ws_size,
                              hipStream_t stream) {
    const float* x      = (const float*)d_in[0];
    const int*   batch  = (const int*)d_in[1];
    const int*   stroke = (const int*)d_in[2];
    const float* W_max  = (const float*)d_in[3];
    const float* g_max  = (const float*)d_in[5];
    const float* be_max = (const float*)d_in[6];
    const float* W_sk   = (const float*)d_in[7];
    const float* g_sk   = (const float*)d_in[9];
    const float* be_sk  = (const float*)d_in[10];
    float* out = (float*)d_out;

    char* ws = (char*)d_ws;
    float2*   Wp       = (float2*)  (ws);                 //   128 KB (both matrices)
    float*    partials = (float*)   (ws + (131072));      //     8 MB
    float*    reduced  = (float*)   (ws + ( 8519680));    //     2 KB
    unsigned* g_enc    = (unsigned*)(ws + ( 8521728));    //    32 KB
    unsigned* s_enc    = (unsigned*)(ws + ( 8554496));    //     4 MB
    float*    gtab     = (float*)   (ws + (12748800));    //    32 KB
    float*    stab     = (float*)   (ws + (12781568));    //     4 MB

    const int n_enc = (NG + NS) * CC;                     // 1,056,768
    k_zero<<<(n_enc + 255) / 256, 256, 0, stream>>>(g_enc, NG * CC);
    k_zero<<<(NS * CC + 255) / 256, 256, 0, stream>>>(s_enc, NS * CC);
    k_pack<<<64, 256, 0, stream>>>(W_max, W_sk, Wp);
    k_gemm<<<NN / 128, 256, 0, stream>>>(x, Wp, batch, stroke, partials, g_enc, s_enc);
    k_reduce<<<256, 256, 0, stream>>>(partials, reduced);
    k_finalize<<<n_enc / 256, 256, 0, stream>>>(g_enc, s_enc, reduced,
                                                g_max, be_max, g_sk, be_sk, gtab, stab);
    k_gather<<<(NN * 64) / 256, 256, 0, stream>>>(batch, stroke, gtab, stab, out);
}